// GNNmodule_14680198218392
// MI455X (gfx1250) — compile-verified
//
#include <hip/hip_runtime.h>
#include <math.h>

// ---------------------------------------------------------------------------
// GCN forward: 3x (GCNConv -> BN -> ReLU[except last]) -> Linear -> sigmoid
// Memory-bound scatter/gather pipeline; dense y1[N,16] @ W2[16,4] mapped to
// V_WMMA_F32_16X16X4_F32 (4 chained K=4 steps, fp32 exact).
// ---------------------------------------------------------------------------

typedef __attribute__((ext_vector_type(2))) float v2f;
typedef __attribute__((ext_vector_type(8))) float v8f;

#define BN_EPS 1e-5f

// ---- deg[i] = 1 (self loop) ------------------------------------------------
__global__ void k_init_deg(float* __restrict__ deg, int n) {
    int i = blockIdx.x * blockDim.x + threadIdx.x;
    if (i < n) deg[i] = 1.0f;
}

// ---- deg[col[e]] += ew[e] --------------------------------------------------
__global__ void k_scatter_deg(float* __restrict__ deg,
                              const int* __restrict__ col,
                              const float* __restrict__ ew, int E) {
    int e = blockIdx.x * blockDim.x + threadIdx.x;
    if (e < E) atomicAdd(&deg[col[e]], ew[e]);
}

// ---- layer1: dis = deg^-1/2 ; hs1 = dis*x*W1 ; agg1 = hs1 (self-loop init) -
__global__ void k_layer1_pre(const float* __restrict__ x,
                             const float* __restrict__ W1,   // [1,16]
                             const float* __restrict__ deg,
                             float* __restrict__ dis,
                             float* __restrict__ hs1,
                             float* __restrict__ agg1, int n) {
    int i = blockIdx.x * blockDim.x + threadIdx.x;
    if (i >= n) return;
    float d = 1.0f / sqrtf(deg[i]);   // deg >= 1 always (self loop)
    dis[i] = d;
    float xs = d * x[i];
#pragma unroll
    for (int f = 0; f < 16; ++f) {
        float v = xs * W1[f];
        hs1[i * 16 + f] = v;
        agg1[i * 16 + f] = v;
    }
}

// ---- edge scatter, F=16: thread per (edge, quad of features) ---------------
__global__ void k_scatter16(float* __restrict__ agg1,
                            const float* __restrict__ hs1,
                            const int* __restrict__ row,
                            const int* __restrict__ col,
                            const float* __restrict__ ew, int E) {
    long long t = (long long)blockIdx.x * blockDim.x + threadIdx.x;
    if (t >= 4LL * E) return;
    int e = (int)(t >> 2);
    int q = (int)(t & 3);
    int r = row[e], c = col[e];
    float w = ew[e];
    const float4 hv = *(const float4*)(hs1 + (size_t)r * 16 + q * 4);
    float* dst = agg1 + (size_t)c * 16 + q * 4;
    atomicAdd(dst + 0, w * hv.x);
    atomicAdd(dst + 1, w * hv.y);
    atomicAdd(dst + 2, w * hv.z);
    atomicAdd(dst + 3, w * hv.w);
}

// ---- layer2: y1 = relu(bn1(dis*agg1 + b1)); h2 = y1 @ W2 (WMMA);
//      hs2 = dis*h2; agg2 = hs2.  One wave per 16-node tile. ----------------
__device__ __forceinline__ float bn_relu(float acc, float dm, float b, float g,
                                         float be, float m, float v) {
    float t = (dm * acc + b - m) * rsqrtf(v + BN_EPS) * g + be;
    return fmaxf(t, 0.0f);
}

__global__ void k_layer2_wmma(const float* __restrict__ agg1,
                              const float* __restrict__ dis,
                              const float* __restrict__ b1,
                              const float* __restrict__ g1,
                              const float* __restrict__ be1,
                              const float* __restrict__ m1,
                              const float* __restrict__ v1,
                              const float* __restrict__ W2,  // [16,4]
                              float* __restrict__ hs2,
                              float* __restrict__ agg2, int n) {
    const int lane = threadIdx.x & 31;
    const int tile = blockIdx.x * (blockDim.x >> 5) + (threadIdx.x >> 5);
    const int full_tiles = n >> 4;
    const int rem = n & 15;
    const int total_tiles = full_tiles + (rem ? 1 : 0);
    if (tile >= total_tiles) return;

    const int mrow = lane & 15;       // A row / B column index for this lane
    const int off  = (lane >> 4) * 2; // upper half-wave holds K+2, K+3

    if (tile < full_tiles) {
        // ---------- WMMA path (all 32 lanes active, EXEC = all ones) -------
        const int node_m = tile * 16 + mrow;
        const float dm = dis[node_m];

        v2f a[4];
#pragma unroll
        for (int k = 0; k < 4; ++k) {
            int f0 = 4 * k + off;
            int f1 = f0 + 1;
            a[k].x = bn_relu(agg1[(size_t)node_m * 16 + f0], dm, b1[f0],
                             g1[f0], be1[f0], m1[f0], v1[f0]);
            a[k].y = bn_relu(agg1[(size_t)node_m * 16 + f1], dm, b1[f1],
                             g1[f1], be1[f1], m1[f1], v1[f1]);
        }

        // B = W2 (16x4) zero-padded to 16x16 columns; row-striped across lanes
        v2f b[4];
#pragma unroll
        for (int k = 0; k < 4; ++k) {
            int k0 = 4 * k + off;
            b[k].x = (mrow < 4) ? W2[k0 * 4 + mrow] : 0.0f;
            b[k].y = (mrow < 4) ? W2[(k0 + 1) * 4 + mrow] : 0.0f;
        }

        v8f c = {0.f, 0.f, 0.f, 0.f, 0.f, 0.f, 0.f, 0.f};
#pragma unroll
        for (int k = 0; k < 4; ++k)
            c = __builtin_amdgcn_wmma_f32_16x16x4_f32(
                    false, a[k], false, b[k], (short)0, c, false, false);

        // C layout: VGPR j -> row j (lanes 0-15) / row j+8 (lanes 16-31)
        if (mrow < 4) {
            const int hi = lane >> 4;
#pragma unroll
            for (int j = 0; j < 8; ++j) {
                int node = tile * 16 + j + 8 * hi;
                float val = dis[node] * c[j];
                size_t idx = (size_t)node * 4 + mrow;
                hs2[idx]  = val;
                agg2[idx] = val;
            }
        }
    } else {
        // ---------- scalar tail for N % 16 nodes ---------------------------
        int node = tile * 16 + lane;
        if (lane < rem) {
            float dm = dis[node];
            float y[16];
#pragma unroll
            for (int f = 0; f < 16; ++f)
                y[f] = bn_relu(agg1[(size_t)node * 16 + f], dm, b1[f],
                               g1[f], be1[f], m1[f], v1[f]);
#pragma unroll
            for (int nf = 0; nf < 4; ++nf) {
                float acc = 0.f;
#pragma unroll
                for (int f = 0; f < 16; ++f) acc += y[f] * W2[f * 4 + nf];
                float val = dm * acc;
                size_t idx = (size_t)node * 4 + nf;
                hs2[idx]  = val;
                agg2[idx] = val;
            }
        }
    }
}

// ---- edge scatter, F=4: thread per edge ------------------------------------
__global__ void k_scatter4(float* __restrict__ agg2,
                           const float* __restrict__ hs2,
                           const int* __restrict__ row,
                           const int* __restrict__ col,
                           const float* __restrict__ ew, int E) {
    int e = blockIdx.x * blockDim.x + threadIdx.x;
    if (e >= E) return;
    int r = row[e], c = col[e];
    float w = ew[e];
    const float4 hv = *(const float4*)(hs2 + (size_t)r * 4);
    float* dst = agg2 + (size_t)c * 4;
    atomicAdd(dst + 0, w * hv.x);
    atomicAdd(dst + 1, w * hv.y);
    atomicAdd(dst + 2, w * hv.z);
    atomicAdd(dst + 3, w * hv.w);
}

// ---- layer3: y2 = relu(bn2(dis*agg2 + b2)); hs3 = dis*(y2@W3); agg3 = hs3 --
__global__ void k_layer3_pre(const float* __restrict__ agg2,
                             const float* __restrict__ dis,
                             const float* __restrict__ b2,
                             const float* __restrict__ g2,
                             const float* __restrict__ be2,
                             const float* __restrict__ m2,
                             const float* __restrict__ v2,
                             const float* __restrict__ W3,  // [4,1]
                             float* __restrict__ hs3,
                             float* __restrict__ agg3, int n) {
    int i = blockIdx.x * blockDim.x + threadIdx.x;
    if (i >= n) return;
    float d = dis[i];
    float h = 0.f;
#pragma unroll
    for (int f = 0; f < 4; ++f) {
        float y = bn_relu(agg2[(size_t)i * 4 + f], d, b2[f],
                          g2[f], be2[f], m2[f], v2[f]);
        h += y * W3[f];
    }
    float val = d * h;
    hs3[i]  = val;
    agg3[i] = val;
}

// ---- edge scatter, F=1 -----------------------------------------------------
__global__ void k_scatter1(float* __restrict__ agg3,
                           const float* __restrict__ hs3,
                           const int* __restrict__ row,
                           const int* __restrict__ col,
                           const float* __restrict__ ew, int E) {
    int e = blockIdx.x * blockDim.x + threadIdx.x;
    if (e >= E) return;
    atomicAdd(&agg3[col[e]], ew[e] * hs3[row[e]]);
}

// ---- final: bn3(dis*agg3 + b3) -> linear -> sigmoid ------------------------
__global__ void k_final(const float* __restrict__ agg3,
                        const float* __restrict__ dis,
                        const float* __restrict__ b3,
                        const float* __restrict__ g3,
                        const float* __restrict__ be3,
                        const float* __restrict__ m3,
                        const float* __restrict__ v3,
                        const float* __restrict__ Wl,
                        const float* __restrict__ bl,
                        float* __restrict__ out, int n) {
    int i = blockIdx.x * blockDim.x + threadIdx.x;
    if (i >= n) return;
    float t = dis[i] * agg3[i] + b3[0];
    t = (t - m3[0]) * rsqrtf(v3[0] + BN_EPS) * g3[0] + be3[0];
    float z = t * Wl[0] + bl[0];
    out[i] = 1.0f / (1.0f + expf(-z));
}

// ---------------------------------------------------------------------------
extern "C" void kernel_launch(void* const* d_in, const int* in_sizes, int n_in,
                              void* d_out, int out_size, void* d_ws, size_t ws_size,
                              hipStream_t stream) {
    (void)n_in; (void)out_size; (void)ws_size;
    const int N = in_sizes[0];   // x is [N,1]
    const int E = in_sizes[2];   // edge_weight is [E]

    const float* x   = (const float*)d_in[0];
    const int*   row = (const int*)d_in[1];      // edge_index[0]
    const int*   col = row + E;                  // edge_index[1]
    const float* ew  = (const float*)d_in[2];
    const float* W1  = (const float*)d_in[3];
    const float* b1  = (const float*)d_in[4];
    const float* g1  = (const float*)d_in[5];
    const float* be1 = (const float*)d_in[6];
    const float* m1  = (const float*)d_in[7];
    const float* v1  = (const float*)d_in[8];
    const float* W2  = (const float*)d_in[9];
    const float* b2  = (const float*)d_in[10];
    const float* g2  = (const float*)d_in[11];
    const float* be2 = (const float*)d_in[12];
    const float* m2  = (const float*)d_in[13];
    const float* v2  = (const float*)d_in[14];
    const float* W3  = (const float*)d_in[15];
    const float* b3  = (const float*)d_in[16];
    const float* g3  = (const float*)d_in[17];
    const float* be3 = (const float*)d_in[18];
    const float* m3  = (const float*)d_in[19];
    const float* v3  = (const float*)d_in[20];
    const float* Wl  = (const float*)d_in[21];
    const float* bl  = (const float*)d_in[22];
    float* out = (float*)d_out;

    // workspace partition: 44*N floats (~17.6 MB for N=100000)
    float* ws   = (float*)d_ws;
    float* deg  = ws;
    float* dis  = deg  + (size_t)N;
    float* hs1  = dis  + (size_t)N;
    float* agg1 = hs1  + (size_t)16 * N;
    float* hs2  = agg1 + (size_t)16 * N;
    float* agg2 = hs2  + (size_t)4 * N;
    float* hs3  = agg2 + (size_t)4 * N;
    float* agg3 = hs3  + (size_t)N;

    const int B = 256;
    k_init_deg<<<(N + B - 1) / B, B, 0, stream>>>(deg, N);
    k_scatter_deg<<<(E + B - 1) / B, B, 0, stream>>>(deg, col, ew, E);
    k_layer1_pre<<<(N + B - 1) / B, B, 0, stream>>>(x, W1, deg, dis, hs1, agg1, N);

    long long t16 = 4LL * E;
    k_scatter16<<<(int)((t16 + B - 1) / B), B, 0, stream>>>(agg1, hs1, row, col, ew, E);

    int tiles = (N + 15) / 16;
    k_layer2_wmma<<<(tiles + 7) / 8, 256, 0, stream>>>(agg1, dis, b1, g1, be1, m1, v1,
                                                       W2, hs2, agg2, N);

    k_scatter4<<<(E + B - 1) / B, B, 0, stream>>>(agg2, hs2, row, col, ew, E);
    k_layer3_pre<<<(N + B - 1) / B, B, 0, stream>>>(agg2, dis, b2, g2, be2, m2, v2,
                                                    W3, hs3, agg3, N);
    k_scatter1<<<(E + B - 1) / B, B, 0, stream>>>(agg3, hs3, row, col, ew, E);
    k_final<<<(N + B - 1) / B, B, 0, stream>>>(agg3, dis, b3, g3, be3, m3, v3,
                                               Wl, bl, out, N);
}